// LogicAutoEncoder_9938554323580
// MI455X (gfx1250) — compile-verified
//
#include <hip/hip_runtime.h>
#include <math.h>

// ---------------------------------------------------------------------------
// LogicNetwork on MI455X (gfx1250).
// Memory-bound problem (113 MB moved, ~1 GFLOP -> ~5us at 23.3 TB/s): output
// depends only on the 3^9 = 19683 possible boards, so we precompute the full
// output table with a WMMA f32 16x16x4 GEMM (act[19683x8] @ heads[8x27] +
// bias) and the main kernel becomes a pure coalesced stream (b128 LDS
// staging) + one L2-resident cacheline-aligned 128B gather per element.
// ---------------------------------------------------------------------------

typedef __attribute__((ext_vector_type(2))) float v2f;
typedef __attribute__((ext_vector_type(8))) float v8f;

#define TBL_ROWS   19683   // 3^9 distinct boards
#define TBL_PAD    19712   // 154 blocks * 8 waves * 16 rows (unguarded stores)
#define TBL_STRIDE 32      // pad 27 -> 32 floats: rows are 128B cacheline-aligned

// ---------------------------------------------------------------------------
// Per-board fuzzy-rule activation (act[r] = prod_p max_i exp(-|wm_i - prem|^2))
// ---------------------------------------------------------------------------
__device__ __forceinline__ float rule_act(const float* __restrict__ prem,
                                          const float player[9], int r) {
  float prod = 1.0f;
#pragma unroll
  for (int p = 0; p < 2; ++p) {
    float a = prem[(r * 2 + p) * 2 + 0];
    float b = prem[(r * 2 + p) * 2 + 1];
    float best = 0.0f;
#pragma unroll
    for (int i = 0; i < 9; ++i) {
      float dx = player[i] - a;
      float dy = (float)(i - 4) * 0.25f - b;
      float sim = __expf(-(dx * dx + dy * dy));
      best = fmaxf(best, sim);
    }
    prod *= best;
  }
  return prod;
}

// ---------------------------------------------------------------------------
// Kernel 1: build table[TBL_PAD][32] = act @ heads + bias via V_WMMA_F32_16X16X4.
// One wave owns 16 boards (M=16); K=8 as two K=4 WMMAs; N=27 padded to 2x16.
// Table rows are padded so every wave stores unconditionally (EXEC stays all-1s,
// no exec-mask save/restore around the WMMA results).
// ---------------------------------------------------------------------------
__global__ __launch_bounds__(256) void logic_build_table(
    const float* __restrict__ premises,  // (8,2,2)
    const float* __restrict__ heads,     // (8,27)
    const float* __restrict__ bias,      // (27,)
    float* __restrict__ table)           // (TBL_PAD,32)
{
  __shared__ float hpad[8 * 32];  // heads zero-padded to 8x32
  int t = threadIdx.x;
  {
    int r = t >> 5, n = t & 31;
    hpad[t] = (n < 27) ? heads[r * 27 + n] : 0.0f;
  }
  __syncthreads();

  int wid  = (blockIdx.x * 256 + t) >> 5;  // global wave id
  int lane = t & 31;
  int base = wid * 16;
  int board = base + (lane & 15);          // rows >= TBL_ROWS land in pad area

  // decode board index -> per-cell player value {0, 1, -1}
  float player[9];
  int rem = board;
#pragma unroll
  for (int i = 0; i < 9; ++i) {
    int s = rem % 3; rem /= 3;
    player[i] = (s == 0) ? 0.0f : ((s == 1) ? 1.0f : -1.0f);
  }

  // A-matrix (16x4 f32, ISA 7.12.2): lanes 0-15 hold K=0,1; lanes 16-31 K=2,3.
  // K-tile 0 -> rules 0..3, K-tile 1 -> rules 4..7.
  int rbase = (lane < 16) ? 0 : 2;
  v2f A0 = { rule_act(premises, player, rbase + 0),
             rule_act(premises, player, rbase + 1) };
  v2f A1 = { rule_act(premises, player, rbase + 4),
             rule_act(premises, player, rbase + 5) };

  int krow = (lane < 16) ? 0 : 2;   // B rows striped like C/D rows: v, v+2
  int row0 = base + ((lane < 16) ? 0 : 8);  // D: lanes 0-15 -> M=v, 16-31 -> M=v+8

#pragma unroll
  for (int n0 = 0; n0 < 32; n0 += 16) {
    int n = n0 + (lane & 15);
    v2f B0 = { hpad[(krow + 0) * 32 + n], hpad[(krow + 1) * 32 + n] };
    v2f B1 = { hpad[(krow + 4) * 32 + n], hpad[(krow + 5) * 32 + n] };
    float bv = (n < 27) ? bias[n] : 0.0f;
    v8f c = { bv, bv, bv, bv, bv, bv, bv, bv };  // C[m][n] = bias[n] for all m
    // D = A0*B0 + C ; D = A1*B1 + D   (accumulate over the two K=4 slices)
    v8f d = __builtin_amdgcn_wmma_f32_16x16x4_f32(
        false, A0, false, B0, (short)0, c, false, false);
    d = __builtin_amdgcn_wmma_f32_16x16x4_f32(
        false, A1, false, B1, (short)0, d, false, false);
#pragma unroll
    for (int v = 0; v < 8; ++v)
      table[(long)(row0 + v) * TBL_STRIDE + n] = d[v];
  }
}

// ---------------------------------------------------------------------------
// Kernel 2: streaming apply. Coalesced b128 in/out via LDS staging; one
// aligned 128B table-row gather per element (table is L2-resident: 2.5 MB).
// ---------------------------------------------------------------------------
__global__ __launch_bounds__(256) void logic_apply(
    const float* __restrict__ board,   // (B,9,3) flat
    const float* __restrict__ table,   // (TBL_PAD,32)
    float* __restrict__ out)           // (B,27) flat
{
  __shared__ __align__(16) float sm[256 * 27];  // 27648 B staging buffer
  int t = threadIdx.x;
  long e0 = (long)blockIdx.x * 256;
  const float* inB  = board + e0 * 27;   // block region: 6912 dwords, 16B aligned
  float*       outB = out   + e0 * 27;

  // coalesced vector load of the block's input region: 6 x b128 + 3 x b32
  const float4* in4 = (const float4*)inB;
  float4* sm4 = (float4*)sm;
#pragma unroll
  for (int q = 0; q < 6; ++q) sm4[q * 256 + t] = in4[q * 256 + t];
#pragma unroll
  for (int q = 0; q < 3; ++q) sm[6144 + q * 256 + t] = inB[6144 + q * 256 + t];
  __syncthreads();

  // per-thread decode (stride-27 LDS reads: gcd(27,64)=1 -> conflict-free)
  int idx = 0, p3 = 1;
#pragma unroll
  for (int i = 0; i < 9; ++i) {
    float c0 = sm[t * 27 + 3 * i + 0];
    float c1 = sm[t * 27 + 3 * i + 1];
    int s = (c0 > 0.5f) ? 0 : ((c1 > 0.5f) ? 1 : 2);
    idx += s * p3;
    p3 *= 3;
  }
  __syncthreads();

  // gather one cacheline-aligned 128B row (28 floats; last is pad)
  const float4* rowp = (const float4*)(table + (long)idx * TBL_STRIDE);
  float4 r[7];
#pragma unroll
  for (int q = 0; q < 7; ++q) r[q] = rowp[q];
  const float* rv = (const float*)r;

#pragma unroll
  for (int j = 0; j < 27; ++j) sm[t * 27 + j] = rv[j];
  __syncthreads();

  // coalesced vector store of the block's output region
  float4* out4 = (float4*)outB;
#pragma unroll
  for (int q = 0; q < 6; ++q) out4[q * 256 + t] = sm4[q * 256 + t];
#pragma unroll
  for (int q = 0; q < 3; ++q) outB[6144 + q * 256 + t] = sm[6144 + q * 256 + t];
}

// ---------------------------------------------------------------------------
// Fallback / tail kernel: fully direct per-element computation.
// ---------------------------------------------------------------------------
__global__ void logic_direct(const float* __restrict__ board,
                             const float* __restrict__ premises,
                             const float* __restrict__ heads,
                             const float* __restrict__ bias,
                             float* __restrict__ out,
                             long nB, long e0)
{
  long e = e0 + (long)blockIdx.x * blockDim.x + threadIdx.x;
  if (e >= nB) return;
  const float* in = board + e * 27;
  float player[9];
#pragma unroll
  for (int i = 0; i < 9; ++i) {
    float c0 = in[3 * i + 0];
    float c1 = in[3 * i + 1];
    player[i] = (c0 > 0.5f) ? 0.0f : ((c1 > 0.5f) ? 1.0f : -1.0f);
  }
  float act[8];
#pragma unroll
  for (int r = 0; r < 8; ++r) act[r] = rule_act(premises, player, r);
#pragma unroll
  for (int n = 0; n < 27; ++n) {
    float acc = bias[n];
#pragma unroll
    for (int r = 0; r < 8; ++r) acc = fmaf(act[r], heads[r * 27 + n], acc);
    out[e * 27 + n] = acc;
  }
}

// ---------------------------------------------------------------------------
extern "C" void kernel_launch(void* const* d_in, const int* in_sizes, int n_in,
                              void* d_out, int out_size, void* d_ws, size_t ws_size,
                              hipStream_t stream) {
  const float* board = (const float*)d_in[0];
  const float* prem  = (const float*)d_in[1];
  const float* heads = (const float*)d_in[2];
  const float* bias  = (const float*)d_in[3];
  float* out = (float*)d_out;

  long nB = (long)in_sizes[0] / 27;                       // 524288
  size_t needed = (size_t)TBL_PAD * TBL_STRIDE * sizeof(float);  // ~2.5 MB

  if (ws_size >= needed) {
    int waves  = (TBL_ROWS + 15) / 16;   // 1231 M-tiles
    int blocks = (waves + 7) / 8;        // 154 blocks of 8 waves
    logic_build_table<<<blocks, 256, 0, stream>>>(prem, heads, bias, (float*)d_ws);

    long nFull = nB & ~255L;
    if (nFull > 0)
      logic_apply<<<(int)(nFull / 256), 256, 0, stream>>>(
          board, (const float*)d_ws, out);
    if (nB > nFull) {
      long rem = nB - nFull;
      logic_direct<<<(int)((rem + 255) / 256), 256, 0, stream>>>(
          board, prem, heads, bias, out, nB, nFull);
    }
  } else {
    logic_direct<<<(int)((nB + 255) / 256), 256, 0, stream>>>(
        board, prem, heads, bias, out, nB, 0);
  }
}